// KSStrong_19224273617028
// MI455X (gfx1250) — compile-verified
//
#include <hip/hip_runtime.h>

// Karplus-Strong as a linear dynamical system on CDNA5 WMMA f32 cores.
// Epoch of dp steps == b <- M b (M is dp x dp, bidiagonal + wrap term);
// samples of epoch e are the state b_e itself. 64 parallel chains seeded with
// A^p b0 (A = M^E) turn the 480k-step scan into E ~ 105..209 sequential
// D x D x 64 GEMM steps executed as v_wmma_f32_16x16x4_f32 tiles, with the
// constant M operand fragments held in registers across the whole chain.

typedef __attribute__((ext_vector_type(2))) float v2f;
typedef __attribute__((ext_vector_type(8))) float v8f;

#define NSTR 4
#define DMAX 80
#define CH   64                  // parallel chains per string (4 WMMA col-tiles)
#define MAT  (DMAX * DMAX)       // 6400 floats per padded matrix slot

// workspace layout (float offsets)
#define MOFF 0
#define AOFF (4 * MAT)
#define TOFF (8 * MAT)
#define SOFF (12 * MAT)                    // seeds: 4 * DMAX*CH
#define POFF (12 * MAT + 4 * DMAX * CH)    // partial outputs: 4 * n

__device__ __forceinline__ int dp_of(int s) {
  const int d[4] = {72, 58, 48, 36};       // int(16000 // f)
  return d[s];
}
__device__ __forceinline__ int D_of(int s) {
  const int d[4] = {80, 64, 48, 48};       // dp padded up to multiple of 16
  return d[s];
}

// One 16x16 f32 tile of R = P(row-major,ldP) x Q(row-major,ldQ), runtime K.
// A-tile (16x4 f32): lanes 0-15 hold K=0,1; lanes 16-31 hold K=2,3.
// B-tile (4x16 f32): mirrored halves. C/D: VGPR v = row v (+8 for hi half).
__device__ __forceinline__ v8f wmma_tile(const float* __restrict__ P, int ldP,
                                         const float* __restrict__ Q, int ldQ,
                                         int rowBase, int colBase, int K, v8f acc) {
  const int lane = threadIdx.x & 31;
  const int half = lane >> 4;
  const int m    = lane & 15;
  for (int k = 0; k < K; k += 4) {
    const int kk = k + 2 * half;
    v2f a, b;
    a.x = P[(rowBase + m) * ldP + kk];
    a.y = P[(rowBase + m) * ldP + kk + 1];
    b.x = Q[kk * ldQ + colBase + m];
    b.y = Q[(kk + 1) * ldQ + colBase + m];
    acc = __builtin_amdgcn_wmma_f32_16x16x4_f32(false, a, false, b,
                                                (short)0, acc, false, false);
  }
  return acc;
}

__device__ __forceinline__ void store_tile(float* __restrict__ R, int ldR,
                                           int rowBase, int colBase, v8f acc) {
  const int lane = threadIdx.x & 31;
  const int half = lane >> 4;
  const int m    = lane & 15;
#pragma unroll
  for (int v = 0; v < 8; ++v)
    R[(rowBase + v + 8 * half) * ldR + colBase + m] = acc[v];
}

// Block-wide DxDxD matmul: Rlds = P x Q (tiles strided over waves, uniform).
__device__ __forceinline__ void matmul_block(float* __restrict__ Rlds,
                                             const float* __restrict__ P,
                                             const float* __restrict__ Q, int D) {
  const int wid = threadIdx.x >> 5;
  const int nw  = blockDim.x >> 5;
  const int nt16 = D / 16;
  const int nt = nt16 * nt16;
  for (int t = wid; t < nt; t += nw) {
    const int rb = (t / nt16) * 16;
    const int cb = (t % nt16) * 16;
    v8f acc = {0.f, 0.f, 0.f, 0.f, 0.f, 0.f, 0.f, 0.f};
    acc = wmma_tile(P, D, Q, D, rb, cb, D, acc);
    store_tile(Rlds, D, rb, cb, acc);
  }
}

// ---- Phase 1: build per-string epoch transition matrix M (padded DxD) ----
__global__ void ks_build(const float* __restrict__ fr, float* __restrict__ M_base) {
  const int s = blockIdx.x;
  const int D = D_of(s), dp = dp_of(s);
  float* M = M_base + s * MAT;
  for (int i = threadIdx.x; i < D * D; i += blockDim.x) M[i] = 0.f;
  __threadfence_block();
  __syncthreads();
  if (threadIdx.x == 0) {
    const float f = fr[s];
    const float gamma = powf(4.0f / logf(f), 1.0f / (float)dp);
    const float g2 = 0.5f * gamma;
    for (int j = 0; j < dp - 1; ++j) {
      M[j * D + j]     = g2;
      M[j * D + j + 1] = g2;
    }
    // last row reads the already-updated b[0]: g2*b[dp-1] + g2^2*(b[0]+b[1])
    M[(dp - 1) * D + (dp - 1)] = g2;
    M[(dp - 1) * D + 0] = g2 * g2;
    M[(dp - 1) * D + 1] = g2 * g2;
  }
}

// ---- Phase 2: A = M^E via binary exponentiation (WMMA matmuls) ----
__global__ void ks_power(const float* __restrict__ M_base, float* __restrict__ A_base,
                         float* __restrict__ T_base, int n) {
  __shared__ float R[MAT];
  const int s = blockIdx.x;
  const int D = D_of(s), dp = dp_of(s);
  const float* M = M_base + s * MAT;
  float* A = A_base + s * MAT;
  float* T = T_base + s * MAT;

  for (int i = threadIdx.x; i < D * D; i += blockDim.x) {
    A[i] = ((i / D) == (i % D)) ? 1.f : 0.f;   // identity
    T[i] = M[i];
  }
  __threadfence_block();
  __syncthreads();

  const int Tep = (n + dp - 1) / dp;
  int e = (Tep + CH - 1) / CH;                 // E = epochs per chain
  while (e > 0) {
    if (e & 1) {
      matmul_block(R, A, T, D);
      __syncthreads();
      for (int i = threadIdx.x; i < D * D; i += blockDim.x) A[i] = R[i];
      __threadfence_block();
      __syncthreads();
    }
    e >>= 1;
    if (e > 0) {
      matmul_block(R, T, T, D);
      __syncthreads();
      for (int i = threadIdx.x; i < D * D; i += blockDim.x) T[i] = R[i];
      __threadfence_block();
      __syncthreads();
    }
  }
}

// ---- Phase 3: seeds S[:,p] = A^p b0 (CH short sequential matvecs) ----
__global__ void ks_seeds(const float* __restrict__ A_base,
                         const float* __restrict__ buf0, const float* __restrict__ buf1,
                         const float* __restrict__ buf2, const float* __restrict__ buf3,
                         float* __restrict__ S_base) {
  __shared__ float x[DMAX], y[DMAX];
  const int s = blockIdx.x;
  const int D = D_of(s), dp = dp_of(s);
  const float* A = A_base + s * MAT;
  float* S = S_base + s * DMAX * CH;
  const float* b0 = (s == 0) ? buf0 : (s == 1) ? buf1 : (s == 2) ? buf2 : buf3;
  const int t = threadIdx.x;

  if (t < D) x[t] = (t < dp) ? b0[t] : 0.f;
  __syncthreads();
  for (int p = 0; p < CH; ++p) {
    if (t < D) {
      S[t * CH + p] = x[t];
      float acc = 0.f;
      for (int k = 0; k < D; ++k) acc = fmaf(A[t * D + k], x[k], acc);
      y[t] = acc;
    }
    __syncthreads();
    if (t < D) x[t] = y[t];
    __syncthreads();
  }
}

// ---- Phase 4: main chain loop, fully specialized per string ----
// X(DxCH) <- M X each epoch; A-fragments of M live in registers for the
// whole chain; only B comes from LDS (ping-ponged X/Y), one barrier/epoch.
template <int D, int DP>
__device__ __forceinline__ void ks_run_body(const float* __restrict__ Mg,
                                            const float* __restrict__ Sg,
                                            float vv, float* __restrict__ part,
                                            float* __restrict__ Xa,
                                            float* __restrict__ Xb, int n) {
  constexpr int NT  = (D / 16) * (CH / 16);   // 16x16 output tiles (1 per wave)
  constexpr int KST = D / 4;                  // WMMA k-steps per tile
  const int tid  = threadIdx.x;
  const int wid  = tid >> 5;
  const int lane = tid & 31;
  const int half = lane >> 4;
  const int m    = lane & 15;
  const int rb   = (wid / (CH / 16)) * 16;
  const int cb   = (wid % (CH / 16)) * 16;

  for (int i = tid; i < D * CH; i += blockDim.x) Xa[i] = Sg[i];

  // Preload this wave's constant A-operand fragments of M into registers.
  v2f afrag[KST];
  if (wid < NT) {
#pragma unroll
    for (int j = 0; j < KST; ++j) {
      const int kk = 4 * j + 2 * half;
      afrag[j].x = Mg[(rb + m) * D + kk];
      afrag[j].y = Mg[(rb + m) * D + kk + 1];
    }
  }
  __syncthreads();

  const int Tep = (n + DP - 1) / DP;
  const int E   = (Tep + CH - 1) / CH;

  float* X = Xa;
  float* Y = Xb;
  for (int e = 0; e < E; ++e) {
    // emit this epoch's samples: sample[(p*E+e)*DP + r] = X[r][p] * vol
    for (int i = tid; i < DP * CH; i += blockDim.x) {
      const int r = i / CH, p = i % CH;
      const int idx = (p * E + e) * DP + r;
      if (idx < n) part[idx] = X[r * CH + p] * vv;
    }
    // Y = M * X  (wave-uniform guard keeps EXEC all-ones around WMMA)
    if (wid < NT) {
      v8f acc = {0.f, 0.f, 0.f, 0.f, 0.f, 0.f, 0.f, 0.f};
#pragma unroll
      for (int j = 0; j < KST; ++j) {
        const int kk = 4 * j + 2 * half;
        v2f b;
        b.x = X[kk * CH + cb + m];
        b.y = X[(kk + 1) * CH + cb + m];
        acc = __builtin_amdgcn_wmma_f32_16x16x4_f32(false, afrag[j], false, b,
                                                    (short)0, acc, false, false);
      }
#pragma unroll
      for (int v = 0; v < 8; ++v)
        Y[(rb + v + 8 * half) * CH + cb + m] = acc[v];
    }
    __syncthreads();
    float* t2 = X; X = Y; Y = t2;
  }
}

__global__ __launch_bounds__(640) void ks_run_all(const float* __restrict__ M_base,
                                                  const float* __restrict__ S_base,
                                                  const float* __restrict__ vol,
                                                  float* __restrict__ part_base,
                                                  int n) {
  __shared__ float Xa[DMAX * CH];   // 20.5 KB
  __shared__ float Xb[DMAX * CH];   // 20.5 KB
  const int s = blockIdx.x;
  const float* Mg = M_base + s * MAT;
  const float* Sg = S_base + s * DMAX * CH;
  float* part = part_base + (size_t)s * (size_t)n;
  const float vv = vol[s];
  __builtin_prefetch(Mg + (threadIdx.x << 3), 0, 1);   // global_prefetch: warm M
  __builtin_prefetch(Sg + (threadIdx.x << 3), 0, 1);   // global_prefetch: warm seeds
  switch (s) {
    case 0:  ks_run_body<80, 72>(Mg, Sg, vv, part, Xa, Xb, n); break;
    case 1:  ks_run_body<64, 58>(Mg, Sg, vv, part, Xa, Xb, n); break;
    case 2:  ks_run_body<48, 48>(Mg, Sg, vv, part, Xa, Xb, n); break;
    default: ks_run_body<48, 36>(Mg, Sg, vv, part, Xa, Xb, n); break;
  }
}

// ---- Phase 5: deterministic reduction of 4 partial streams ----
__global__ void ks_sum(const float* __restrict__ part, float* __restrict__ out, int n) {
  const int n4 = n >> 2;
  const int i = blockIdx.x * blockDim.x + threadIdx.x;
  const float4* p0 = (const float4*)(part);
  const float4* p1 = (const float4*)(part + (size_t)n);
  const float4* p2 = (const float4*)(part + 2 * (size_t)n);
  const float4* p3 = (const float4*)(part + 3 * (size_t)n);
  float4* o4 = (float4*)out;
  if (i < n4) {
    float4 a = p0[i], b = p1[i], c = p2[i], d = p3[i];
    float4 r;
    r.x = a.x + b.x + c.x + d.x;
    r.y = a.y + b.y + c.y + d.y;
    r.z = a.z + b.z + c.z + d.z;
    r.w = a.w + b.w + c.w + d.w;
    o4[i] = r;
  }
  if (i == 0) {
    for (int j = n4 << 2; j < n; ++j)
      out[j] = part[j] + part[(size_t)n + j] + part[2 * (size_t)n + j] +
               part[3 * (size_t)n + j];
  }
}

extern "C" void kernel_launch(void* const* d_in, const int* in_sizes, int n_in,
                              void* d_out, int out_size, void* d_ws, size_t ws_size,
                              hipStream_t stream) {
  (void)in_sizes; (void)n_in; (void)ws_size;
  const float* fr  = (const float*)d_in[0];
  const float* vol = (const float*)d_in[1];
  const float* b0  = (const float*)d_in[2];
  const float* b1  = (const float*)d_in[3];
  const float* b2  = (const float*)d_in[4];
  const float* b3  = (const float*)d_in[5];
  // d_in[6] = n_samples on device; host-side n == out_size.
  const int n = out_size;

  float* ws   = (float*)d_ws;
  float* M    = ws + MOFF;
  float* A    = ws + AOFF;
  float* T    = ws + TOFF;
  float* S    = ws + SOFF;
  float* part = ws + POFF;
  float* out  = (float*)d_out;

  ks_build<<<NSTR, 256, 0, stream>>>(fr, M);
  ks_power<<<NSTR, 256, 0, stream>>>(M, A, T, n);
  ks_seeds<<<NSTR, 128, 0, stream>>>(A, b0, b1, b2, b3, S);
  ks_run_all<<<NSTR, 640, 0, stream>>>(M, S, vol, part, n);
  const int n4 = n >> 2;
  const int blocks = (n4 + 255) / 256 > 0 ? (n4 + 255) / 256 : 1;
  ks_sum<<<blocks, 256, 0, stream>>>(part, out, n);
}